// ResidualBKPConv_14568529068256
// MI455X (gfx1250) — compile-verified
//
#include <hip/hip_runtime.h>

// Problem constants (from reference)
#define N_PTS   100000
#define NS_PTS  12500
#define N_EDGE  400000
#define F_IN    64
#define HID     32
#define F_OUT   128
#define KPTS    16
// KP_EXTENT = 1.0/1.5 ; 1/extent^2 = 2.25
#define INV_EXT2 2.25f

typedef __attribute__((ext_vector_type(2))) float v2f;
typedef __attribute__((ext_vector_type(8))) float v8f;

// D(16x16,f32) = A(16x4,f32) x B(4x16,f32) + C
// args: (neg_a, A, neg_b, B, c_mod, C, reuse_a, reuse_b)
static __device__ __forceinline__ v8f wmma4(v2f a, v2f b, v8f c) {
  return __builtin_amdgcn_wmma_f32_16x16x4_f32(false, a, false, b, (short)0, c,
                                               false, false);
}

// ---------------------------------------------------------------------------
// Stage 1: x_side[N,32] = x[N,64] @ pre_W[64,32] + pre_b     (N % 16 == 0)
// One wave per 16-row tile; 2 column tiles (HID=32).
// ---------------------------------------------------------------------------
__global__ __launch_bounds__(128)
void pre_gemm(const float* __restrict__ x, const float* __restrict__ W,
              const float* __restrict__ bias, float* __restrict__ y) {
  const int tile = blockIdx.x * 4 + (threadIdx.x >> 5);
  if (tile >= N_PTS / 16) return;
  const int lane = threadIdx.x & 31;
  const int half = lane >> 4, l16 = lane & 15;
  const int row0 = tile * 16;
  const int m = row0 + l16;                      // A row for this lane
  v8f c0 = {}, c1 = {};
  for (int k0 = 0; k0 < F_IN; k0 += 4) {
    const int ka = k0 + 2 * half;
    v2f a;  a.x = x[(size_t)m * F_IN + ka];  a.y = x[(size_t)m * F_IN + ka + 1];
    v2f b0; b0.x = W[ka * HID + l16];        b0.y = W[(ka + 1) * HID + l16];
    v2f b1; b1.x = W[ka * HID + 16 + l16];   b1.y = W[(ka + 1) * HID + 16 + l16];
    c0 = wmma4(a, b0, c0);
    c1 = wmma4(a, b1, c1);
  }
  const float bv0 = bias[l16], bv1 = bias[16 + l16];
  for (int v = 0; v < 8; ++v) {
    const int r = row0 + v + 8 * half;
    y[(size_t)r * HID + l16]      = c0[v] + bv0;
    y[(size_t)r * HID + 16 + l16] = c1[v] + bv1;
  }
}

// ---------------------------------------------------------------------------
// Stage 2: zero the per-(dst, kernel-point) bins (atomics accumulate into it)
// ---------------------------------------------------------------------------
__global__ void zero_f32(float* __restrict__ p, size_t n) {
  size_t i = (size_t)blockIdx.x * blockDim.x + threadIdx.x;
  const size_t stride = (size_t)gridDim.x * blockDim.x;
  for (; i < n; i += stride) p[i] = 0.0f;
}

// ---------------------------------------------------------------------------
// Stage 3: per-edge KP selection + binned scatter.
// One wave32 per edge, lane == feature channel (HID == 32 == wave width).
// tmp[dst][nn][c] += w * x_side[src][c]
// ---------------------------------------------------------------------------
__global__ __launch_bounds__(256)
void edge_scatter(const float* __restrict__ pos, const int* __restrict__ idx,
                  const int* __restrict__ esrc, const int* __restrict__ edst,
                  const float* __restrict__ kern, const float* __restrict__ xs,
                  float* __restrict__ tmp) {
  const int e = (int)((blockIdx.x * blockDim.x + threadIdx.x) >> 5);
  if (e >= N_EDGE) return;
  const int lane = threadIdx.x & 31;
  const int src = esrc[e];
  const int dst = edst[e];
  const int sub = idx[dst];
  const float rx = pos[(size_t)src * 3 + 0] - pos[(size_t)sub * 3 + 0];
  const float ry = pos[(size_t)src * 3 + 1] - pos[(size_t)sub * 3 + 1];
  const float rz = pos[(size_t)src * 3 + 2] - pos[(size_t)sub * 3 + 2];
  float best = 3.4e38f;
  int nn = 0;
  for (int k = 0; k < KPTS; ++k) {        // redundant across lanes, trivial VALU
    const float dx = rx - kern[k * 3 + 0];
    const float dy = ry - kern[k * 3 + 1];
    const float dz = rz - kern[k * 3 + 2];
    const float d2 = dx * dx + dy * dy + dz * dz;
    if (d2 < best) { best = d2; nn = k; } // '<' keeps first min, like argmin
  }
  float w = 1.0f - best * INV_EXT2;
  w = w > 0.0f ? w : 0.0f;
  const float v = w * xs[(size_t)src * HID + lane];
  atomicAdd(&tmp[(((size_t)dst * KPTS + nn) * HID) + lane], v);
}

// ---------------------------------------------------------------------------
// Stage 4: agg[NS,32] = sum_k tmp[:,k,:] @ kernel_weight[k]   (WMMA, K=32 x16)
// ---------------------------------------------------------------------------
__global__ __launch_bounds__(128)
void agg_gemm(const float* __restrict__ tmp, const float* __restrict__ kw,
              float* __restrict__ agg) {
  const int tiles = (NS_PTS + 15) / 16;
  const int tile = blockIdx.x * 4 + (threadIdx.x >> 5);
  if (tile >= tiles) return;
  const int lane = threadIdx.x & 31;
  const int half = lane >> 4, l16 = lane & 15;
  const int row0 = tile * 16;
  const int m = row0 + l16;
  const bool mok = (m < NS_PTS);
  v8f c0 = {}, c1 = {};
  for (int k = 0; k < KPTS; ++k) {
    const float* __restrict__ A = tmp + ((size_t)m * KPTS + k) * HID;
    const float* __restrict__ B = kw + (size_t)k * HID * HID;
    for (int kk0 = 0; kk0 < HID; kk0 += 4) {
      const int ka = kk0 + 2 * half;
      v2f a;
      a.x = mok ? A[ka]     : 0.0f;
      a.y = mok ? A[ka + 1] : 0.0f;
      v2f b0; b0.x = B[ka * HID + l16];      b0.y = B[(ka + 1) * HID + l16];
      v2f b1; b1.x = B[ka * HID + 16 + l16]; b1.y = B[(ka + 1) * HID + 16 + l16];
      c0 = wmma4(a, b0, c0);
      c1 = wmma4(a, b1, c1);
    }
  }
  for (int v = 0; v < 8; ++v) {
    const int r = row0 + v + 8 * half;
    if (r < NS_PTS) {
      agg[(size_t)r * HID + l16]      = c0[v];
      agg[(size_t)r * HID + 16 + l16] = c1[v];
    }
  }
}

// ---------------------------------------------------------------------------
// Stage 5: out[NS,128] = agg @ post_W + x[idx] @ short_W + post_b + short_b
// grid.x: row tiles (x4 waves), grid.y: 8 column tiles of 16.
// ---------------------------------------------------------------------------
__global__ __launch_bounds__(128)
void out_gemm(const float* __restrict__ agg, const float* __restrict__ postW,
              const float* __restrict__ postb, const float* __restrict__ x,
              const int* __restrict__ idx, const float* __restrict__ shortW,
              const float* __restrict__ shortb, float* __restrict__ out) {
  const int tiles = (NS_PTS + 15) / 16;
  const int tile = blockIdx.x * 4 + (threadIdx.x >> 5);
  if (tile >= tiles) return;
  const int col0 = blockIdx.y * 16;
  const int lane = threadIdx.x & 31;
  const int half = lane >> 4, l16 = lane & 15;
  const int row0 = tile * 16;
  const int m = row0 + l16;
  const bool mok = (m < NS_PTS);
  const int gi = mok ? idx[m] : 0;
  v8f c = {};
  // agg[NS,32] @ post_W[32,128]
  for (int k0 = 0; k0 < HID; k0 += 4) {
    const int ka = k0 + 2 * half;
    v2f a;
    a.x = mok ? agg[(size_t)m * HID + ka]     : 0.0f;
    a.y = mok ? agg[(size_t)m * HID + ka + 1] : 0.0f;
    v2f b;
    b.x = postW[ka * F_OUT + col0 + l16];
    b.y = postW[(ka + 1) * F_OUT + col0 + l16];
    c = wmma4(a, b, c);
  }
  // x[idx][NS,64] @ short_W[64,128]  (shortcut path, fused into same accum)
  for (int k0 = 0; k0 < F_IN; k0 += 4) {
    const int ka = k0 + 2 * half;
    v2f a;
    a.x = mok ? x[(size_t)gi * F_IN + ka]     : 0.0f;
    a.y = mok ? x[(size_t)gi * F_IN + ka + 1] : 0.0f;
    v2f b;
    b.x = shortW[ka * F_OUT + col0 + l16];
    b.y = shortW[(ka + 1) * F_OUT + col0 + l16];
    c = wmma4(a, b, c);
  }
  const float bv = postb[col0 + l16] + shortb[col0 + l16];
  for (int v = 0; v < 8; ++v) {
    const int r = row0 + v + 8 * half;
    if (r < NS_PTS)
      out[(size_t)r * F_OUT + col0 + l16] = c[v] + bv;
  }
}

// ---------------------------------------------------------------------------
extern "C" void kernel_launch(void* const* d_in, const int* in_sizes, int n_in,
                              void* d_out, int out_size, void* d_ws, size_t ws_size,
                              hipStream_t stream) {
  // setup_inputs() order:
  const float* x       = (const float*)d_in[0];   // [N,64]
  const float* pos     = (const float*)d_in[1];   // [N,3]
  const float* pre_W   = (const float*)d_in[2];   // [64,32]
  const float* pre_b   = (const float*)d_in[3];   // [32]
  const float* kern    = (const float*)d_in[4];   // [16,3]
  const float* kern_W  = (const float*)d_in[5];   // [16,32,32]
  const float* post_W  = (const float*)d_in[6];   // [32,128]
  const float* post_b  = (const float*)d_in[7];   // [128]
  const float* short_W = (const float*)d_in[8];   // [64,128]
  const float* short_b = (const float*)d_in[9];   // [128]
  const int*   idx     = (const int*)d_in[10];    // [NS]
  const int*   esrc    = (const int*)d_in[11];    // [E]
  const int*   edst    = (const int*)d_in[12];    // [E]
  float* out = (float*)d_out;                     // [NS,128]

  // workspace layout (floats)
  float* xs  = (float*)d_ws;                              // [N,32]
  float* tmp = xs + (size_t)N_PTS * HID;                  // [NS,16,32]
  float* agg = tmp + (size_t)NS_PTS * KPTS * HID;         // [NS,32]

  // 1) pre MLP (WMMA f32)
  pre_gemm<<<dim3((N_PTS / 16 + 3) / 4), dim3(128), 0, stream>>>(x, pre_W, pre_b, xs);

  // 2) zero bins
  zero_f32<<<dim3(1024), dim3(256), 0, stream>>>(tmp, (size_t)NS_PTS * KPTS * HID);

  // 3) per-edge KP select + binned scatter (one wave32 per edge)
  edge_scatter<<<dim3(N_EDGE / 8), dim3(256), 0, stream>>>(pos, idx, esrc, edst,
                                                           kern, xs, tmp);

  // 4) aggregate via 16 small dense GEMMs (WMMA f32)
  const int tiles = (NS_PTS + 15) / 16;
  agg_gemm<<<dim3((tiles + 3) / 4), dim3(128), 0, stream>>>(tmp, kern_W, agg);

  // 5) fused post-MLP + shortcut GEMM (WMMA f32)
  out_gemm<<<dim3((tiles + 3) / 4, F_OUT / 16), dim3(128), 0, stream>>>(
      agg, post_W, post_b, x, idx, short_W, short_b, out);
}